// GraphTransformer_2817498546754
// MI455X (gfx1250) — compile-verified
//
#include <hip/hip_runtime.h>
#include <math.h>

typedef __attribute__((ext_vector_type(2))) float v2f;
typedef __attribute__((ext_vector_type(8))) float v8f;

#define FD 64  // feature dim

// ---------------------------------------------------------------------------
// WMMA GEMM: out[M x 64] = A[M x 64] @ W[64 x 64] (+ bias) (optional relu)
// One wave computes one 16x16 output tile via 16 chained V_WMMA_F32_16X16X4_F32.
// Block = 128 threads = 4 waves = one full 16-row x 64-col stripe.
// ---------------------------------------------------------------------------
__global__ __launch_bounds__(128)
void gemm64_wmma(const float* __restrict__ A, const float* __restrict__ W,
                 const float* __restrict__ bias, float* __restrict__ out,
                 int M, int relu)
{
    const int lane  = threadIdx.x & 31;
    const int wave  = threadIdx.x >> 5;      // 0..3 -> column tile
    const int r0    = blockIdx.x * 16;       // row tile base
    const int n0    = wave * 16;             // col tile base
    const int lm    = lane & 15;             // M (for A) / N (for B) within tile
    const int khalf = (lane >> 4) * 2;       // K sub-offset: lanes 0-15 -> 0, 16-31 -> 2

    int mrow = r0 + lm;
    if (mrow >= M) mrow = M - 1;             // clamp tail loads; stores guarded below
    const float* __restrict__ arow = A + (size_t)mrow * FD;

    v8f c = {};
#pragma unroll
    for (int k0 = 0; k0 < FD; k0 += 4) {
        v2f a, b;
        // A 16x4 fp32 layout: VGPR0 = K=khalf, VGPR1 = K=khalf+1 (ISA 7.12.2)
        a.x = arow[k0 + khalf + 0];
        a.y = arow[k0 + khalf + 1];
        // B 4x16 fp32 layout (rows striped across lanes, mirrored from A)
        b.x = W[(size_t)(k0 + khalf + 0) * FD + n0 + lm];
        b.y = W[(size_t)(k0 + khalf + 1) * FD + n0 + lm];
        c = __builtin_amdgcn_wmma_f32_16x16x4_f32(false, a, false, b,
                                                  (short)0, c, false, false);
    }

    const int col = n0 + lm;
    const float bv = bias ? bias[col] : 0.0f;
#pragma unroll
    for (int r = 0; r < 8; ++r) {
        int row = r0 + r + ((lane >> 4) << 3);   // C/D: VGPR r -> M=r (lo lanes), M=r+8 (hi)
        if (row < M) {
            float vv = c[r] + bv;
            if (relu) vv = fmaxf(vv, 0.0f);
            out[(size_t)row * FD + col] = vv;
        }
    }
}

// ---------------------------------------------------------------------------
// Degree / normalization helpers
// ---------------------------------------------------------------------------
__global__ void fill_ones(float* p, int n) {
    int i = blockIdx.x * blockDim.x + threadIdx.x;
    if (i < n) p[i] = 1.0f;                         // self-loop contributes 1
}

__global__ void count_deg(const int* __restrict__ dst, float* deg, int E) {
    int e = blockIdx.x * blockDim.x + threadIdx.x;
    if (e < E) atomicAdd(&deg[dst[e]], 1.0f);
}

__global__ void rsqrt_inplace(float* p, int n) {
    int i = blockIdx.x * blockDim.x + threadIdx.x;
    if (i < n) p[i] = rsqrtf(p[i]);                 // deg >= 1 always (self-loop)
}

// ---------------------------------------------------------------------------
// GCN scatter: out[n] = xw[n]*dinv[n]^2 (+bias) then += xw[s]*dinv[s]*dinv[d]
// ---------------------------------------------------------------------------
__global__ void gcn_init(const float* __restrict__ xw, const float* __restrict__ dinv,
                         const float* __restrict__ bias, float* __restrict__ out, int n)
{
    int i = blockIdx.x * blockDim.x + threadIdx.x;     // over n * 64
    if (i < n * FD) {
        int node = i >> 6, t = i & 63;
        float dv = dinv[node];
        float v  = xw[i] * dv * dv;
        if (bias) v += bias[t];
        out[i] = v;
    }
}

__global__ void gcn_scatter(const float* __restrict__ xw, const float* __restrict__ dinv,
                            const int* __restrict__ src, const int* __restrict__ dst,
                            float* __restrict__ out, int E)
{
    long long tid = (long long)blockIdx.x * blockDim.x + threadIdx.x;  // over E * 64
    int e = (int)(tid >> 6);
    int t = (int)(tid & 63);
    if (e < E) {
        int s = src[e], d = dst[e];
        float nrm = dinv[s] * dinv[d];
        atomicAdd(&out[(size_t)d * FD + t], xw[(size_t)s * FD + t] * nrm);
    }
}

__global__ void bias_relu_inplace(float* __restrict__ x, const float* __restrict__ bias, int total) {
    int i = blockIdx.x * blockDim.x + threadIdx.x;
    if (i < total) x[i] = fmaxf(x[i] + bias[i & 63], 0.0f);
}

// ---------------------------------------------------------------------------
// TransformerConv pieces
// ---------------------------------------------------------------------------
__device__ __forceinline__ void atomicMaxFloat(float* addr, float val) {
    // works for values initialized to -inf: int-max for >=0, uint-min for <0
    if (val >= 0.0f) atomicMax((int*)addr, __float_as_int(val));
    else             atomicMin((unsigned int*)addr, __float_as_uint(val));
}

__global__ void init_m_z(float* m, float* z, int n) {
    int i = blockIdx.x * blockDim.x + threadIdx.x;
    if (i < n) { m[i] = -INFINITY; z[i] = 0.0f; }
}

// one wave per edge: logits[e] = dot(q[dst], k[src]) / 8 ; atomic max into m[dst]
__global__ __launch_bounds__(256)
void edge_logits(const float* __restrict__ q, const float* __restrict__ k,
                 const int* __restrict__ src, const int* __restrict__ dst,
                 float* __restrict__ logits, float* __restrict__ m, int E)
{
    int e    = blockIdx.x * 8 + (threadIdx.x >> 5);
    int lane = threadIdx.x & 31;
    if (e >= E) return;
    int s = src[e], d = dst[e];
    const float2* qp = (const float2*)(q + (size_t)d * FD);
    const float2* kp = (const float2*)(k + (size_t)s * FD);
    float2 qa = qp[lane];
    float2 ka = kp[lane];
    float p = qa.x * ka.x + qa.y * ka.y;
#pragma unroll
    for (int off = 16; off > 0; off >>= 1) p += __shfl_xor(p, off, 32);
    if (lane == 0) {
        float lg = p * 0.125f;                        // rsqrt(64)
        logits[e] = lg;
        atomicMaxFloat(&m[d], lg);
    }
}

// logits[e] <- exp(logits[e] - m[dst]); z[dst] += exp
__global__ void edge_exp_z(float* __restrict__ logits, const float* __restrict__ m,
                           const int* __restrict__ dst, float* __restrict__ z, int E)
{
    int e = blockIdx.x * blockDim.x + threadIdx.x;
    if (e < E) {
        int d = dst[e];
        float ex = expf(logits[e] - m[d]);
        logits[e] = ex;
        atomicAdd(&z[d], ex);
    }
}

// out[d] += v[s] * (expo[e] / z[d])   (out pre-loaded with skip connection)
__global__ void attn_scatter(const float* __restrict__ vmat, const float* __restrict__ expo,
                             const float* __restrict__ z,
                             const int* __restrict__ src, const int* __restrict__ dst,
                             float* __restrict__ out, int E)
{
    long long tid = (long long)blockIdx.x * blockDim.x + threadIdx.x;  // over E * 64
    int e = (int)(tid >> 6);
    int t = (int)(tid & 63);
    if (e < E) {
        int s = src[e], d = dst[e];
        float alpha = expo[e] / z[d];                 // z[d] > 0 whenever edge e exists
        atomicAdd(&out[(size_t)d * FD + t], vmat[(size_t)s * FD + t] * alpha);
    }
}

// ---------------------------------------------------------------------------
// Host-side orchestration
// ---------------------------------------------------------------------------
extern "C" void kernel_launch(void* const* d_in, const int* in_sizes, int n_in,
                              void* d_out, int out_size, void* d_ws, size_t ws_size,
                              hipStream_t stream)
{
    const float* x   = (const float*)d_in[0];
    const int*   ei  = (const int*)  d_in[1];
    const float* W1  = (const float*)d_in[2];
    const float* b1  = (const float*)d_in[3];
    const float* Wq  = (const float*)d_in[4];
    const float* bq  = (const float*)d_in[5];
    const float* Wk  = (const float*)d_in[6];
    const float* bk  = (const float*)d_in[7];
    const float* Wv  = (const float*)d_in[8];
    const float* bv  = (const float*)d_in[9];
    const float* Ws  = (const float*)d_in[10];
    const float* bs  = (const float*)d_in[11];
    const float* W2  = (const float*)d_in[12];
    const float* b2  = (const float*)d_in[13];
    float* out       = (float*)d_out;

    const int N = in_sizes[0] / FD;
    const int E = in_sizes[1] / 2;
    const int* src = ei;
    const int* dst = ei + E;

    // workspace layout (floats)
    float* ws    = (float*)d_ws;
    size_t nd    = (size_t)N * FD;
    float* xw    = ws;                 // [N,64] GEMM output (reused for GCN2)
    float* x1    = ws + 1 * nd;        // [N,64]
    float* qb    = ws + 2 * nd;        // [N,64]
    float* kb    = ws + 3 * nd;        // [N,64]
    float* vb    = ws + 4 * nd;        // [N,64]
    float* x2    = ws + 5 * nd;        // [N,64] skip + attention aggregate
    float* dinv  = ws + 6 * nd;        // [N] deg -> rsqrt(deg)
    float* mmax  = dinv + N;           // [N]
    float* zsum  = mmax + N;           // [N]
    float* lgs   = zsum + N;           // [E] logits -> exp(logits - m)

    const int TB = 256;
    dim3 blk(TB);
    dim3 gN((N + TB - 1) / TB);
    dim3 gE((E + TB - 1) / TB);
    dim3 gNF(((int)nd + TB - 1) / TB);
    dim3 gEF((int)(((long long)E * FD + TB - 1) / TB));
    dim3 gEwave((E + 7) / 8);
    dim3 gGemm((N + 15) / 16);
    dim3 gemmBlk(128);

    // ---- degree + symmetric normalization -------------------------------
    fill_ones   <<<gN, blk, 0, stream>>>(dinv, N);
    count_deg   <<<gE, blk, 0, stream>>>(dst, dinv, E);
    rsqrt_inplace<<<gN, blk, 0, stream>>>(dinv, N);

    // ---- GCN layer 1: x1 = relu(GCN(x, W1) + b1) ------------------------
    gemm64_wmma <<<gGemm, gemmBlk, 0, stream>>>(x, W1, nullptr, xw, N, 0);
    gcn_init    <<<gNF, blk, 0, stream>>>(xw, dinv, nullptr, x1, N);
    gcn_scatter <<<gEF, blk, 0, stream>>>(xw, dinv, src, dst, x1, E);
    bias_relu_inplace<<<gNF, blk, 0, stream>>>(x1, b1, (int)nd);

    // ---- TransformerConv: q/k/v/skip GEMMs ------------------------------
    gemm64_wmma <<<gGemm, gemmBlk, 0, stream>>>(x1, Wq, bq, qb, N, 0);
    gemm64_wmma <<<gGemm, gemmBlk, 0, stream>>>(x1, Wk, bk, kb, N, 0);
    gemm64_wmma <<<gGemm, gemmBlk, 0, stream>>>(x1, Wv, bv, vb, N, 0);
    gemm64_wmma <<<gGemm, gemmBlk, 0, stream>>>(x1, Ws, bs, x2, N, 0);  // skip into x2

    // ---- segment softmax over edges, aggregate into x2 ------------------
    init_m_z    <<<gN, blk, 0, stream>>>(mmax, zsum, N);
    edge_logits <<<gEwave, blk, 0, stream>>>(qb, kb, src, dst, lgs, mmax, E);
    edge_exp_z  <<<gE, blk, 0, stream>>>(lgs, mmax, dst, zsum, E);
    attn_scatter<<<gEF, blk, 0, stream>>>(vb, lgs, zsum, src, dst, x2, E);

    // ---- GCN layer 2: out = GCN(x2, W2) + b2 ----------------------------
    gemm64_wmma <<<gGemm, gemmBlk, 0, stream>>>(x2, W2, nullptr, xw, N, 0);
    gcn_init    <<<gNF, blk, 0, stream>>>(xw, dinv, b2, out, N);
    gcn_scatter <<<gEF, blk, 0, stream>>>(xw, dinv, src, dst, out, E);
}